// ObjectCondensationLoss_13855564497089
// MI455X (gfx1250) — compile-verified
//
#include <hip/hip_runtime.h>
#include <math.h>

#define KCOLS 256        // 255 real object columns + 1 dead pad column
#define EPSF  1e-9f

typedef __attribute__((ext_vector_type(2))) float v2f;
typedef __attribute__((ext_vector_type(8))) float v8f;

struct Accum {
    float att_sum, rep_sum, noise_sum, ce_sum;
    float mw_sum, mh_sum, me_sum, mp_sum;
    float coward_sum;
    unsigned att_cnt, rep_cnt, noise_cnt, denom_cnt, valid_cnt;
};

__device__ __forceinline__ float block_reduce_256(float v, float* sbuf) {
    int t = threadIdx.x;
    sbuf[t] = v;
    __syncthreads();
    for (int s = 128; s > 0; s >>= 1) {
        if (t < s) sbuf[t] += sbuf[t + s];
        __syncthreads();
    }
    float r = sbuf[0];
    __syncthreads();
    return r;
}

__device__ __forceinline__ float finite_or_zero(float v) {
    return (isnan(v) || isinf(v)) ? 0.0f : v;
}

// ---------------- Kernel 0: zero workspace state ----------------
__global__ void k0_init(unsigned long long* __restrict__ obj_best, Accum* __restrict__ acc) {
    int t = threadIdx.x;
    if (t < KCOLS) obj_best[t] = 0ull;
    if (t == 0) {
        acc->att_sum = 0.f; acc->rep_sum = 0.f; acc->noise_sum = 0.f; acc->ce_sum = 0.f;
        acc->mw_sum = 0.f; acc->mh_sum = 0.f; acc->me_sum = 0.f; acc->mp_sum = 0.f;
        acc->coward_sum = 0.f;
        acc->att_cnt = 0u; acc->rep_cnt = 0u; acc->noise_cnt = 0u;
        acc->denom_cnt = 0u; acc->valid_cnt = 0u;
    }
}

// ---------------- Kernel 1: per-hit pass ----------------
__global__ __launch_bounds__(256) void k1_perhit(
    const float* __restrict__ beta, const float* __restrict__ x,
    const int* __restrict__ oid, const float* __restrict__ logits,
    const int* __restrict__ tgt,
    const float* __restrict__ wd, const float* __restrict__ wdp,
    const float* __restrict__ ht, const float* __restrict__ htp,
    const float* __restrict__ en, const float* __restrict__ enp,
    const float* __restrict__ mo, const float* __restrict__ mop,
    float* __restrict__ q, float* __restrict__ xx,
    unsigned long long* __restrict__ obj_best, Accum* __restrict__ acc, int N)
{
    __shared__ float sbuf[256];
    int i = blockIdx.x * 256 + threadIdx.x;

    float noise_s = 0.f, ce_s = 0.f, mw = 0.f, mh = 0.f, me = 0.f, mp = 0.f;
    float noise_c = 0.f, den_c = 0.f;

    if (i < N) {
        float b  = beta[i];
        float th = atanhf(b);
        float qi = th * th + 1.0f;
        q[i] = qi;

        const float* xr = x + (size_t)i * 8;
        float s = 0.f;
        #pragma unroll
        for (int d = 0; d < 8; ++d) s = fmaf(xr[d], xr[d], s);
        xx[i] = s;

        int o = oid[i];
        if (o > 0) {
            // pack (q_bits, ~i): q >= 1 so float bits are positive & order-preserving;
            // ~i makes atomicMax pick the LOWEST index on ties, like jnp.argmax.
            unsigned long long key =
                ((unsigned long long)__float_as_uint(qi) << 32) |
                (unsigned long long)(unsigned)(~(unsigned)i);
            atomicMax(&obj_best[o - 1], key);
            den_c = 1.0f;

            const float* lr = logits + (size_t)i * 8;
            float mx = lr[0];
            #pragma unroll
            for (int c = 1; c < 8; ++c) mx = fmaxf(mx, lr[c]);
            float se = 0.f;
            #pragma unroll
            for (int c = 0; c < 8; ++c) se += expf(lr[c] - mx);
            int tg = tgt[i];
            ce_s = mx + logf(se) - lr[tg];

            float d0 = wd[i] - wdp[i]; mw = d0 * d0;
            float d1 = ht[i] - htp[i]; mh = d1 * d1;
            float d2 = en[i] - enp[i]; me = d2 * d2;
            float d3 = mo[i] - mop[i]; mp = d3 * d3;
        } else {
            noise_s = b;
            noise_c = 1.0f;
        }
    }

    float r;
    r = block_reduce_256(noise_s, sbuf); if (threadIdx.x == 0) atomicAdd(&acc->noise_sum, r);
    r = block_reduce_256(ce_s,    sbuf); if (threadIdx.x == 0) atomicAdd(&acc->ce_sum, r);
    r = block_reduce_256(mw,      sbuf); if (threadIdx.x == 0) atomicAdd(&acc->mw_sum, r);
    r = block_reduce_256(mh,      sbuf); if (threadIdx.x == 0) atomicAdd(&acc->mh_sum, r);
    r = block_reduce_256(me,      sbuf); if (threadIdx.x == 0) atomicAdd(&acc->me_sum, r);
    r = block_reduce_256(mp,      sbuf); if (threadIdx.x == 0) atomicAdd(&acc->mp_sum, r);
    r = block_reduce_256(noise_c, sbuf); if (threadIdx.x == 0) atomicAdd(&acc->noise_cnt, (unsigned)(r + 0.5f));
    r = block_reduce_256(den_c,   sbuf); if (threadIdx.x == 0) atomicAdd(&acc->denom_cnt, (unsigned)(r + 0.5f));
}

// ---------------- Kernel 2: gather condensation points ----------------
__global__ __launch_bounds__(256) void k2_gather(
    const float* __restrict__ beta, const float* __restrict__ x,
    const float* __restrict__ xx,
    const unsigned long long* __restrict__ obj_best,
    float* __restrict__ xa, float* __restrict__ qa,
    float* __restrict__ xxa, float* __restrict__ validf,
    Accum* __restrict__ acc, int N)
{
    __shared__ float sbuf[256];
    int k = threadIdx.x;           // single block of 256

    float cow = 0.f, vc = 0.f;
    float qv = 0.f, xxv = 0.f, vf = 0.f;
    float xrow[8] = {0.f, 0.f, 0.f, 0.f, 0.f, 0.f, 0.f, 0.f};

    if (k < KCOLS - 1) {
        unsigned long long key = obj_best[k];
        if (key != 0ull) {
            int alpha = (int)(~(unsigned)(key & 0xffffffffull));
            qv  = __uint_as_float((unsigned)(key >> 32));
            xxv = xx[alpha];
            vf  = 1.0f;
            const float* xr = x + (size_t)alpha * 8;
            #pragma unroll
            for (int d = 0; d < 8; ++d) xrow[d] = xr[d];
            cow = 1.0f - beta[alpha];
            vc  = 1.0f;
        }
    }
    qa[k] = qv; xxa[k] = xxv; validf[k] = vf;
    #pragma unroll
    for (int d = 0; d < 8; ++d) xa[k * 8 + d] = xrow[d];

    float r;
    r = block_reduce_256(cow, sbuf); if (k == 0) atomicAdd(&acc->coward_sum, r);
    r = block_reduce_256(vc,  sbuf); if (k == 0) atomicAdd(&acc->valid_cnt, (unsigned)(r + 0.5f));
}

// ---------------- Kernel 3: N x 256 pair loss via WMMA ----------------
__global__ __launch_bounds__(256) void k3_pairs(
    const float* __restrict__ x, const float* __restrict__ q,
    const float* __restrict__ xx, const int* __restrict__ oid,
    const float* __restrict__ xa, const float* __restrict__ qa,
    const float* __restrict__ xxa, const float* __restrict__ validf,
    Accum* __restrict__ acc, int N, int NT)
{
    __shared__ float s_xa[KCOLS * 8];
    __shared__ float s_qa[KCOLS];
    __shared__ float s_xxa[KCOLS];
    __shared__ float s_val[KCOLS];
    __shared__ float sbuf[256];

    int t = threadIdx.x;
    for (int idx = t; idx < KCOLS * 8; idx += 256) s_xa[idx] = xa[idx];
    if (t < KCOLS) { s_qa[t] = qa[t]; s_xxa[t] = xxa[t]; s_val[t] = validf[t]; }
    __syncthreads();

    int wave = blockIdx.x * 8 + (t >> 5);
    int lane = t & 31;

    float att_s = 0.f, rep_s = 0.f, att_c = 0.f, rep_c = 0.f;

    if (wave < NT) {                       // wave-uniform: EXEC stays all-1s for WMMA
        int row0 = wave * 16;
        int half = lane >> 4;              // 0: K 0..1 / rows M 0..7 ; 1: K 2..3 / rows M 8..15
        int l15  = lane & 15;
        int ka   = half * 2;

        // A fragment (16x4 f32): lane l15 holds row row0+l15, K = ka, ka+1
        int arow  = row0 + l15;
        int arowc = arow < N ? arow : N - 1;
        const float* xr = x + (size_t)arowc * 8;
        v2f A0; A0.x = xr[ka];     A0.y = xr[ka + 1];
        v2f A1; A1.x = xr[4 + ka]; A1.y = xr[4 + ka + 1];

        // per-row values for the C-layout rows this lane owns (M = r + 8*half)
        int rbase = row0 + half * 8;
        float qrow[8], xxrow[8]; int orow[8];
        #pragma unroll
        for (int r = 0; r < 8; ++r) {
            int rr = rbase + r;
            int rc = rr < N ? rr : N - 1;
            qrow[r]  = q[rc];
            xxrow[r] = xx[rc];
            orow[r]  = (rr < N) ? oid[rc] : -1;   // -1 never matches a column id
        }

        for (int ct = 0; ct < KCOLS / 16; ++ct) {
            int c = ct * 16 + l15;
            // B fragment (4x16 f32): lane l15 holds column c, K = ka, ka+1
            v2f B0; B0.x = s_xa[c * 8 + ka];     B0.y = s_xa[c * 8 + ka + 1];
            v2f B1; B1.x = s_xa[c * 8 + 4 + ka]; B1.y = s_xa[c * 8 + 4 + ka + 1];

            v8f C = {0.f, 0.f, 0.f, 0.f, 0.f, 0.f, 0.f, 0.f};
            C = __builtin_amdgcn_wmma_f32_16x16x4_f32(false, A0, false, B0, (short)0, C, false, false);
            C = __builtin_amdgcn_wmma_f32_16x16x4_f32(false, A1, false, B1, (short)0, C, false, false);

            float qac  = s_qa[c];
            float xxac = s_xxa[c];
            bool  vcol = s_val[c] != 0.0f;
            int   ocol = c + 1;                    // column c <-> object id c+1

            union { v8f v; float f[8]; } u; u.v = C;
            #pragma unroll
            for (int r = 0; r < 8; ++r) {
                float d2 = xxrow[r] + xxac - 2.0f * u.f[r];
                d2 = fmaxf(d2, 0.0f);
                bool att  = (orow[r] == ocol);
                float qw  = qrow[r] * qac;
                bool a_ok = att && vcol;
                att_s += a_ok ? qw * d2 : 0.0f;
                att_c += a_ok ? 1.0f : 0.0f;
                float dist = sqrtf(fmaxf(d2, 1e-12f));
                bool r_ok = (!att) && vcol && (dist < 1.0f) && (orow[r] >= 0);
                rep_s += r_ok ? qw * (1.0f - dist) : 0.0f;
                rep_c += r_ok ? 1.0f : 0.0f;
            }
        }
    }

    float r;
    r = block_reduce_256(att_s, sbuf); if (t == 0) atomicAdd(&acc->att_sum, r);
    r = block_reduce_256(rep_s, sbuf); if (t == 0) atomicAdd(&acc->rep_sum, r);
    r = block_reduce_256(att_c, sbuf); if (t == 0) atomicAdd(&acc->att_cnt, (unsigned)(r + 0.5f));
    r = block_reduce_256(rep_c, sbuf); if (t == 0) atomicAdd(&acc->rep_cnt, (unsigned)(r + 0.5f));
}

// ---------------- Kernel 4: finalize 12 outputs ----------------
__global__ void k4_final(const Accum* __restrict__ acc, float* __restrict__ out) {
    if (threadIdx.x != 0 || blockIdx.x != 0) return;
    float v_att  = acc->att_sum   / ((float)acc->att_cnt   + EPSF);
    float n_rep  = (float)acc->rep_cnt;
    float v_rep  = acc->rep_sum   / (n_rep                 + EPSF);
    float l_cow  = acc->coward_sum/ ((float)acc->valid_cnt + EPSF);
    float l_noi  = acc->noise_sum / ((float)acc->noise_cnt + EPSF);
    float oc = finite_or_zero(v_att) + finite_or_zero(v_rep)
             + finite_or_zero(l_cow) + finite_or_zero(l_noi);
    float denom = (float)acc->denom_cnt;
    float ce = acc->ce_sum / denom;
    float mw = acc->mw_sum / denom;
    float mh = acc->mh_sum / denom;
    float me = acc->me_sum / denom;
    float mp = acc->mp_sum / denom;
    float total = oc + ce + 0.1f * mw + 0.1f * mh + me + mp;
    out[0]  = v_att; out[1] = v_rep; out[2] = l_cow; out[3] = l_noi; out[4] = n_rep;
    out[5]  = oc;    out[6] = ce;    out[7] = mw;    out[8] = mh;
    out[9]  = me;    out[10] = mp;   out[11] = total;
}

extern "C" void kernel_launch(void* const* d_in, const int* in_sizes, int n_in,
                              void* d_out, int out_size, void* d_ws, size_t ws_size,
                              hipStream_t stream)
{
    const float* beta   = (const float*)d_in[0];
    const float* x      = (const float*)d_in[1];
    const int*   oid    = (const int*)  d_in[2];
    const float* logits = (const float*)d_in[3];
    const int*   tgt    = (const int*)  d_in[4];
    const float* wd     = (const float*)d_in[5];
    const float* wdp    = (const float*)d_in[6];
    const float* ht     = (const float*)d_in[7];
    const float* htp    = (const float*)d_in[8];
    const float* en     = (const float*)d_in[9];
    const float* enp    = (const float*)d_in[10];
    const float* mo     = (const float*)d_in[11];
    const float* mop    = (const float*)d_in[12];
    const int N = in_sizes[0];

    char* w = (char*)d_ws;
    size_t off = 0;
    auto take = [&](size_t bytes) -> void* {
        void* p = (void*)(w + off);
        off = (off + bytes + 255) & ~(size_t)255;
        return p;
    };
    float*              q        = (float*)take((size_t)N * 4);
    float*              xx       = (float*)take((size_t)N * 4);
    unsigned long long* obj_best = (unsigned long long*)take(KCOLS * 8);
    float*              xa       = (float*)take(KCOLS * 8 * 4);
    float*              qa       = (float*)take(KCOLS * 4);
    float*              xxa      = (float*)take(KCOLS * 4);
    float*              validf   = (float*)take(KCOLS * 4);
    Accum*              acc      = (Accum*)take(sizeof(Accum));

    k0_init<<<1, 256, 0, stream>>>(obj_best, acc);

    int blocks1 = (N + 255) / 256;
    k1_perhit<<<blocks1, 256, 0, stream>>>(beta, x, oid, logits, tgt,
                                           wd, wdp, ht, htp, en, enp, mo, mop,
                                           q, xx, obj_best, acc, N);

    k2_gather<<<1, 256, 0, stream>>>(beta, x, xx, obj_best, xa, qa, xxa, validf, acc, N);

    int NT = (N + 15) / 16;
    int blocks3 = (NT + 7) / 8;
    k3_pairs<<<blocks3, 256, 0, stream>>>(x, q, xx, oid, xa, qa, xxa, validf, acc, N, NT);

    k4_final<<<1, 1, 0, stream>>>(acc, (float*)d_out);
}